// RQSBatch_52810917871889
// MI455X (gfx1250) — compile-verified
//
#include <hip/hip_runtime.h>
#include <hip/hip_bf16.h>
#include <stdint.h>

// ---------------- problem constants (match reference) ----------------
#define NUMF   2048
#define NBIN   16
#define NBATCH 16384
// BOUND=5, MIN_DERIVATIVE=0.001, MIN_BIN_WIDTH=MIN_BIN_HEIGHT=0.001

// ---------------- per-feature record layout in ws / LDS --------------
// invw[16] | h[16] | s[16] | d[17] | kx[17] | ky[17] | pad -> 104 floats (416B, 16B-mult)
#define REC    104
#define O_INVW 0
#define O_H    16
#define O_S    32
#define O_D    48
#define O_KX   65
#define O_KY   82

#define TF       128                 // features per block tile
#define GRID_Y   128                 // batch tiles
#define ROWS_PB  (NBATCH / GRID_Y)   // 128 rows per block
#define LDS_DW   (TF * REC)          // 13312 dwords = 53248 bytes

typedef float        f32x4 __attribute__((ext_vector_type(4)));
typedef unsigned int uv4   __attribute__((ext_vector_type(4)));
typedef int          iv4   __attribute__((ext_vector_type(4)));
typedef int          iv8   __attribute__((ext_vector_type(8)));

// =====================================================================
// Kernel 1: per-feature spline parameter tables -> workspace
// (2048 features; one thread per feature; trivial cost)
// =====================================================================
__global__ __launch_bounds__(256) void rqs_params_kernel(
    const float* __restrict__ uw, const float* __restrict__ uh,
    const float* __restrict__ ud, float* __restrict__ ws)
{
  const int f = blockIdx.x * 256 + threadIdx.x;
  if (f >= NUMF) return;

  float w[NBIN], h[NBIN], d[NBIN + 1];

  // softmax(widths) -> (0.001 + 0.984*sm) * 10
  {
    const float* p = uw + (size_t)f * NBIN;
    float m = -3.0e38f;
#pragma unroll
    for (int i = 0; i < NBIN; ++i) { w[i] = p[i]; m = fmaxf(m, w[i]); }
    float sum = 0.f;
#pragma unroll
    for (int i = 0; i < NBIN; ++i) { w[i] = expf(w[i] - m); sum += w[i]; }
    const float inv = 1.0f / sum;
#pragma unroll
    for (int i = 0; i < NBIN; ++i) w[i] = (0.001f + 0.984f * (w[i] * inv)) * 10.0f;
  }
  // softmax(heights) -> same
  {
    const float* p = uh + (size_t)f * NBIN;
    float m = -3.0e38f;
#pragma unroll
    for (int i = 0; i < NBIN; ++i) { h[i] = p[i]; m = fmaxf(m, h[i]); }
    float sum = 0.f;
#pragma unroll
    for (int i = 0; i < NBIN; ++i) { h[i] = expf(h[i] - m); sum += h[i]; }
    const float inv = 1.0f / sum;
#pragma unroll
    for (int i = 0; i < NBIN; ++i) h[i] = (0.001f + 0.984f * (h[i] * inv)) * 10.0f;
  }
  // softplus(derivatives) + 0.001   (stable: max(x,0)+log1p(exp(-|x|)))
  {
    const float* p = ud + (size_t)f * (NBIN + 1);
#pragma unroll
    for (int i = 0; i <= NBIN; ++i) {
      const float v = p[i];
      d[i] = fmaxf(v, 0.0f) + log1pf(expf(-fabsf(v))) + 0.001f;
    }
  }

  float* rec = ws + (size_t)f * REC;
  float kx = -5.0f, ky = -5.0f;
  rec[O_KX] = kx;
  rec[O_KY] = ky;
#pragma unroll
  for (int i = 0; i < NBIN; ++i) {
    rec[O_INVW + i] = 1.0f / w[i];
    rec[O_H    + i] = h[i];
    rec[O_S    + i] = h[i] / w[i];
    kx += w[i]; ky += h[i];
    rec[O_KX + i + 1] = kx;
    rec[O_KY + i + 1] = ky;
  }
#pragma unroll
  for (int i = 0; i <= NBIN; ++i) rec[O_D + i] = d[i];
}

// =====================================================================
// Per-element rational-quadratic spline evaluation (tables in LDS)
// =====================================================================
__device__ __forceinline__ float rqs_eval(const float* __restrict__ tb, float xx,
                                          float d0, float dK, float kxK, float kyK)
{
  // idx = #{ j in 1..15 : kx[j] < xx }  (== searchsorted into inner knots, clipped)
  int idx = (xx > tb[O_KX + 8]) ? 8 : 0;
  idx += (xx > tb[O_KX + idx + 4]) ? 4 : 0;
  idx += (xx > tb[O_KX + idx + 2]) ? 2 : 0;
  idx += (xx > tb[O_KX + idx + 1]) ? 1 : 0;

  const float invw = tb[O_INVW + idx];
  const float h    = tb[O_H    + idx];
  const float s    = tb[O_S    + idx];
  const float dk   = tb[O_D    + idx];
  const float dk1  = tb[O_D    + idx + 1];
  const float xk   = tb[O_KX   + idx];
  const float yk   = tb[O_KY   + idx];

  const float xin = fminf(fmaxf(xx, -5.0f), 5.0f);
  const float xi  = (xin - xk) * invw;
  const float tt  = xi * (1.0f - xi);
  const float num = fmaf(s * xi, xi, dk * tt);
  const float den = fmaf(fmaf(-2.0f, s, dk + dk1), tt, s);   // strictly > 0
#if __has_builtin(__builtin_amdgcn_rcpf)
  float r = __builtin_amdgcn_rcpf(den);
  r = r * fmaf(-den, r, 2.0f);                // one Newton step, ~<=1 ulp
#else
  const float r = 1.0f / den;
#endif
  float y = fmaf(h * num, r, yk);
  // linear tails (knot_x[0]=knot_y[0]=-BOUND are constants)
  y = (xx < -5.0f) ? fmaf(d0, xx + 5.0f, -5.0f) : y;
  y = (xx >  5.0f) ? fmaf(dK, xx - kxK, kyK)    : y;
  return y;
}

// =====================================================================
// Kernel 2: main transform.  Block = 256 thr, 128 features x 128 rows.
// Table tile staged into LDS via TDM (tensor_load_to_lds + s_wait_tensorcnt).
// =====================================================================
__global__ __launch_bounds__(256) void rqs_main_kernel(
    const float* __restrict__ x, const float* __restrict__ ws,
    float* __restrict__ out)
{
  __shared__ float tab[LDS_DW];
  const int f0 = blockIdx.x * TF;
  const int r0 = blockIdx.y * ROWS_PB;
  const int t  = threadIdx.x;

#if __has_builtin(__builtin_amdgcn_tensor_load_to_lds)
  if (t < 32) {                 // one wave issues the TDM op (EXEC ignored by TDM)
    const uint64_t ga  = (uint64_t)(uintptr_t)(ws + (size_t)f0 * REC);
    const uint32_t lds = (uint32_t)(uintptr_t)(&tab[0]); // low 32b of flat = LDS offset
    const uint32_t nd  = LDS_DW;                         // dwords in tile (<= 65535)

    uv4 g0; iv8 g1;
    // --- D# group 0 (CDNA5 ISA) ---
    g0[0] = 1u;                                           // count=1, no gather/restore
    g0[1] = lds;                                          // lds_addr
    g0[2] = (uint32_t)ga;                                 // global_addr[31:0]
    g0[3] = ((uint32_t)(ga >> 32) & 0x01FFFFFFu) | (2u << 30); // addr[56:32] | type=2
    // --- D# group 1 (§8.4): 1-D copy, data_size=4B ---
    g1[0] = (int)(2u << 16);                              // data_size=2 (4 bytes)
    g1[1] = (int)((nd & 0xFFFFu) << 16);                  // tensor_dim0[15:0]
    g1[2] = (int)(((nd >> 16) & 0xFFFFu) | (1u << 16));   // tensor_dim0[31:16] | tensor_dim1=1
    g1[3] = (int)((nd & 0xFFFFu) << 16);                  // tensor_dim1_hi=0 | tile_dim0=nd
    g1[4] = (int)1;                                       // tile_dim1=1, tile_dim2=0
    g1[5] = (int)nd;                                      // tensor_dim0_stride[31:0]
    g1[6] = 0;                                            // stride0_hi | dim1_stride lo
    g1[7] = 0;
    const iv4 g2 = {0, 0, 0, 0};                          // <=2-D tensor: groups 2/3 unused
    const iv4 g3 = {0, 0, 0, 0};
    const iv8 g4 = {0, 0, 0, 0, 0, 0, 0, 0};              // unused trailing group (6-arg form)
    __builtin_amdgcn_tensor_load_to_lds(g0, g1, g2, g3, g4, 0);
#if __has_builtin(__builtin_amdgcn_s_wait_tensorcnt)
    __builtin_amdgcn_s_wait_tensorcnt(0);
#else
    asm volatile("s_wait_tensorcnt 0" ::: "memory");
#endif
  }
#else
  // fallback: plain vector copy global -> LDS
  {
    const f32x4* src = (const f32x4*)(ws + (size_t)f0 * REC);
    f32x4* dst = (f32x4*)tab;
    for (int i = t; i < LDS_DW / 4; i += 256) dst[i] = src[i];
  }
#endif
  __syncthreads();

  // thread t -> 4 consecutive features (b128 vectorization), 1 of 8 row phases
  const int fg = t & 31;        // feature group within tile: 0..31
  const int rs = t >> 5;        // row phase: 0..7
  const size_t col = (size_t)f0 + (size_t)fg * 4;

  const float* tb[4];
  float d0[4], dK[4], kxK[4], kyK[4];
#pragma unroll
  for (int j = 0; j < 4; ++j) {
    tb[j]  = &tab[(fg * 4 + j) * REC];
    d0[j]  = tb[j][O_D];
    dK[j]  = tb[j][O_D + NBIN];
    kxK[j] = tb[j][O_KX + NBIN];
    kyK[j] = tb[j][O_KY + NBIN];
  }

  for (int r = r0 + rs; r < r0 + ROWS_PB; r += 8) {
    const float* xp = x + (size_t)r * NUMF + col;
    if (r + 8 < r0 + ROWS_PB)
      __builtin_prefetch(xp + (size_t)8 * NUMF, 0, 0);    // global_prefetch_b8
    const f32x4 xv = __builtin_nontemporal_load((const f32x4*)xp);
    f32x4 ov;
    ov.x = rqs_eval(tb[0], xv.x, d0[0], dK[0], kxK[0], kyK[0]);
    ov.y = rqs_eval(tb[1], xv.y, d0[1], dK[1], kxK[1], kyK[1]);
    ov.z = rqs_eval(tb[2], xv.z, d0[2], dK[2], kxK[2], kyK[2]);
    ov.w = rqs_eval(tb[3], xv.w, d0[3], dK[3], kxK[3], kyK[3]);
    __builtin_nontemporal_store(ov, (f32x4*)(out + (size_t)r * NUMF + col));
  }
}

// =====================================================================
extern "C" void kernel_launch(void* const* d_in, const int* in_sizes, int n_in,
                              void* d_out, int out_size, void* d_ws, size_t ws_size,
                              hipStream_t stream)
{
  const float* x  = (const float*)d_in[0];
  const float* uw = (const float*)d_in[1];
  const float* uh = (const float*)d_in[2];
  const float* ud = (const float*)d_in[3];
  float* out = (float*)d_out;
  float* ws  = (float*)d_ws;     // needs NUMF*REC*4 = 851,968 bytes

  rqs_params_kernel<<<dim3((NUMF + 255) / 256), dim3(256), 0, stream>>>(uw, uh, ud, ws);
  rqs_main_kernel<<<dim3(NUMF / TF, GRID_Y), dim3(256), 0, stream>>>(x, ws, out);
}